// ARMTTTMultiHeadSelfAttention_49950469652782
// MI455X (gfx1250) — compile-verified
//
#include <hip/hip_runtime.h>
#include <hip/hip_bf16.h>
#include <stdint.h>

typedef __attribute__((ext_vector_type(2))) float v2f;
typedef __attribute__((ext_vector_type(4))) float v4f;
typedef __attribute__((ext_vector_type(8))) float v8f;

// D(16x16,f32) += A(16x4,f32) x B(4x16,f32)
#define WMMA_F32X4(c, a, b)                                                    \
  c = __builtin_amdgcn_wmma_f32_16x16x4_f32(false, a, false, b, (short)0, c,   \
                                            false, false)

// Wait until at most N async(-to-LDS) ops remain outstanding for this wave.
#define WAIT_ASYNC(n) asm volatile("s_wait_asynccnt %0" ::"n"(n) : "memory")

constexpr int kB = 16, kH = 12, kN = 4096, kD = 64;
constexpr int kIT = 16;              // N_ITERS
constexpr int kM = 256;              // chunk rows = N / N_ITERS
constexpr int kWS = 68;              // W^T stride: even (b64 align), ≡4 mod 64
constexpr int kES = 260;             // err^T stride: even, ≡4 mod 64
constexpr int kOutStride = kH * kD;  // 768
constexpr float kInvMD = 1.0f / (float)(kM * kD);  // LR=1 grad scale

// Async-copy one contiguous 64KB chunk (256x64 fp32) global -> LDS.
// 512 threads x 16B = 8KB per issue; 8 issues per thread. ASYNCcnt += 8.
// INST_OFFSET is added to BOTH the LDS and the global address (ISA 15.18.3).
__device__ __forceinline__ void async_copy_chunk(const float* __restrict__ g,
                                                 const float* lds_base,
                                                 int tid) {
  const uint32_t loff = (uint32_t)(uintptr_t)lds_base + (uint32_t)tid * 16u;
  const char* gp = (const char*)g + (size_t)tid * 16;
#pragma unroll
  for (int r = 0; r < 8; ++r) {
    asm volatile("global_load_async_to_lds_b128 %0, %1, off offset:%2" ::"v"(
                     loff),
                 "v"(gp), "n"(r * 8192)
                 : "memory");
  }
}

__launch_bounds__(512, 1)
__global__ void ttt_fastweight_kernel(const float* __restrict__ q,
                                      const float* __restrict__ k,
                                      const float* __restrict__ v,
                                      const float* __restrict__ w0,
                                      float* __restrict__ out) {
  extern __shared__ float smem[];
  float* sWt = smem;                  // 64 x 68   W^T (Wt[col][row])
  float* sK0 = sWt + kD * kWS;        // 256 x 64  Xtr buffer 0 (row-major)
  float* sK1 = sK0 + kM * kD;         // 256 x 64  Xtr buffer 1
  float* sQ  = sK1 + kM * kD;         // 256 x 64  Xte buffer (row-major)
  float* sEt = sQ + kM * kD;          // 64 x 260  err^T (Et[col][row])

  const int tid  = threadIdx.x;
  const int lane = tid & 31;
  const int wave = tid >> 5;          // 0..15
  const int half = lane >> 4;         // fragment half per ISA layout
  const int lm   = lane & 15;

  const int bh = blockIdx.x;          // 0..191
  const int b  = bh / kH;
  const int h  = bh - b * kH;
  const size_t baseBH = (size_t)bh * kN * kD;

  // init W^T from W_init[h]
  for (int t = tid; t < kD * kD; t += blockDim.x)
    sWt[(t & 63) * kWS + (t >> 6)] = w0[h * kD * kD + t];

  // prefetch k-chunk(0)
  async_copy_chunk(k + baseBH, sK0, tid);
  __syncthreads();  // covers sWt init

  for (int it = 0; it < kIT; ++it) {
    const size_t chunkOff = baseBH + (size_t)it * kM * kD;
    const float* sKb = (it & 1) ? sK1 : sK0;

    // prefetch q-chunk(it); consumed only in Phase E (hidden under B+C)
    async_copy_chunk(q + chunkOff, sQ, tid);  // ASYNCcnt +8
    WAIT_ASYNC(8);      // k-chunk(it) complete (in-order retirement)
    __syncthreads();    // all waves' portions visible

    // ---- Phase B: err = Xtr*W - Ytr. Wave owns row-tile strip r=wave;
    //      4 col-tiles share the A fragment (4 independent WMMA chains).
    //      A: contiguous b64 from sKb rows; B: contiguous b64 from sWt cols.
    {
      const int rowBase = wave << 4;
      v8f c0, c1, c2, c3;
      const float* gv = v + chunkOff + (size_t)(rowBase + 8 * half) * kD + lm;
#pragma unroll
      for (int p = 0; p < 8; ++p) {
        c0[p] = -gv[(size_t)p * kD + 0];
        c1[p] = -gv[(size_t)p * kD + 16];
        c2[p] = -gv[(size_t)p * kD + 32];
        c3[p] = -gv[(size_t)p * kD + 48];
      }
      const float* xrow = sKb + (rowBase + lm) * kD + 2 * half;
      const float* wcol = sWt + (size_t)lm * kWS + 2 * half;
#pragma unroll
      for (int kt = 0; kt < 16; ++kt) {
        const int kBase = kt << 2;
        v2f a  = *(const v2f*)&xrow[kBase];
        v2f b0 = *(const v2f*)&wcol[kBase + 0 * 16 * kWS];
        v2f b1 = *(const v2f*)&wcol[kBase + 1 * 16 * kWS];
        v2f b2 = *(const v2f*)&wcol[kBase + 2 * 16 * kWS];
        v2f b3 = *(const v2f*)&wcol[kBase + 3 * 16 * kWS];
        WMMA_F32X4(c0, a, b0);
        WMMA_F32X4(c1, a, b1);
        WMMA_F32X4(c2, a, b2);
        WMMA_F32X4(c3, a, b3);
      }
      // store err transposed: per lane 8 consecutive floats -> 2x b128 per tile
      float* e0 = sEt + (size_t)lm * kES + rowBase + 8 * half;
      v4f lo, hi;
      lo = (v4f){c0[0], c0[1], c0[2], c0[3]};
      hi = (v4f){c0[4], c0[5], c0[6], c0[7]};
      *(v4f*)&e0[0 * 16 * kES] = lo; *(v4f*)&e0[0 * 16 * kES + 4] = hi;
      lo = (v4f){c1[0], c1[1], c1[2], c1[3]};
      hi = (v4f){c1[4], c1[5], c1[6], c1[7]};
      *(v4f*)&e0[1 * 16 * kES] = lo; *(v4f*)&e0[1 * 16 * kES + 4] = hi;
      lo = (v4f){c2[0], c2[1], c2[2], c2[3]};
      hi = (v4f){c2[4], c2[5], c2[6], c2[7]};
      *(v4f*)&e0[2 * 16 * kES] = lo; *(v4f*)&e0[2 * 16 * kES + 4] = hi;
      lo = (v4f){c3[0], c3[1], c3[2], c3[3]};
      hi = (v4f){c3[4], c3[5], c3[6], c3[7]};
      *(v4f*)&e0[3 * 16 * kES] = lo; *(v4f*)&e0[3 * 16 * kES + 4] = hi;
    }
    __syncthreads();

    // ---- Phase C: W^T -= (err^T * Xtr) / (m*D).  (G = Xtr^T err => G^T here.)
    //      A = err^T: contiguous b64 from sEt. B = Xtr: K-strided pair.
    //      1 tile per wave, split-K dual chains. Disjoint W^T tiles.
    {
      const int ibase = (wave >> 2) << 4;  // err-column block (W^T row)
      const int jbase = (wave & 3) << 4;   // Xtr-column block (W^T col)
      v8f c0 = {}, c1 = {};
      const float* arow = sEt + (size_t)(ibase + lm) * kES + 2 * half;
      const float* bcol = sKb + jbase + lm;
#pragma unroll 4
      for (int kk = 0; kk < 64; kk += 2) {
        const int kBase0 = kk << 2;
        const int kBase1 = kBase0 + 4;
        v2f a0 = *(const v2f*)&arow[kBase0];
        v2f a1 = *(const v2f*)&arow[kBase1];
        v2f b0, b1;
        const int r0 = (kBase0 + 2 * half) * kD;
        b0.x = bcol[r0];
        b0.y = bcol[r0 + kD];
        const int r1 = (kBase1 + 2 * half) * kD;
        b1.x = bcol[r1];
        b1.y = bcol[r1 + kD];
        WMMA_F32X4(c0, a0, b0);
        WMMA_F32X4(c1, a1, b1);
      }
#pragma unroll
      for (int p = 0; p < 8; ++p) {
        const int idx = (ibase + p + 8 * half) * kWS + jbase + lm;
        sWt[idx] -= (c0[p] + c1[p]) * kInvMD;
      }
    }
    __syncthreads();

    // prefetch next k-chunk into the other buffer; overlaps with Phase E
    if (it + 1 < kIT) {
      async_copy_chunk(k + chunkOff + (size_t)kM * kD, (it & 1) ? sK0 : sK1,
                       tid);               // ASYNCcnt: q(8) + k(8)
      WAIT_ASYNC(8);                       // q-chunk done, k continues
    } else {
      WAIT_ASYNC(0);                       // only q outstanding
    }
    __syncthreads();

    // ---- Phase E: out = Xte * Wn, same strip decomposition as Phase B.
    {
      const int rowBase = wave << 4;
      v8f c0 = {}, c1 = {}, c2 = {}, c3 = {};
      const float* xrow = sQ + (rowBase + lm) * kD + 2 * half;
      const float* wcol = sWt + (size_t)lm * kWS + 2 * half;
#pragma unroll
      for (int kt = 0; kt < 16; ++kt) {
        const int kBase = kt << 2;
        v2f a  = *(const v2f*)&xrow[kBase];
        v2f b0 = *(const v2f*)&wcol[kBase + 0 * 16 * kWS];
        v2f b1 = *(const v2f*)&wcol[kBase + 1 * 16 * kWS];
        v2f b2 = *(const v2f*)&wcol[kBase + 2 * 16 * kWS];
        v2f b3 = *(const v2f*)&wcol[kBase + 3 * 16 * kWS];
        WMMA_F32X4(c0, a, b0);
        WMMA_F32X4(c1, a, b1);
        WMMA_F32X4(c2, a, b2);
        WMMA_F32X4(c3, a, b3);
      }
      float* go = out +
                  ((size_t)b * kN + (size_t)it * kM + rowBase + 8 * half) *
                      kOutStride +
                  h * kD + lm;
#pragma unroll
      for (int p = 0; p < 8; ++p) {
        go[(size_t)p * kOutStride + 0]  = c0[p];
        go[(size_t)p * kOutStride + 16] = c1[p];
        go[(size_t)p * kOutStride + 32] = c2[p];
        go[(size_t)p * kOutStride + 48] = c3[p];
      }
    }
    __syncthreads();  // sQ free for next iteration's prefetch
  }
}

extern "C" void kernel_launch(void* const* d_in, const int* in_sizes, int n_in,
                              void* d_out, int out_size, void* d_ws,
                              size_t ws_size, hipStream_t stream) {
  const float* q  = (const float*)d_in[0];
  const float* k  = (const float*)d_in[1];
  const float* v  = (const float*)d_in[2];
  const float* w0 = (const float*)d_in[3];
  float* out = (float*)d_out;

  const size_t shmem =
      (size_t)(kD * kWS + 3 * kM * kD + kD * kES) * sizeof(float);  // ~274 KB
  dim3 grid(kB * kH);  // 192 independent (b,h) fast-weight loops
  dim3 block(512);     // 16 waves of 32
  ttt_fastweight_kernel<<<grid, block, shmem, stream>>>(q, k, v, w0, out);
}